// FeatureVoxel_15693810499597
// MI455X (gfx1250) — compile-verified
//
#include <hip/hip_runtime.h>
#include <hip/hip_bf16.h>
#include <stdint.h>

#define RES 256
#define CH  16
// voxel layout: (257, 257, 257, 16) float32, strides in floats:
#define ZS  (16)
#define YS  (257 * 16)           // 4112
#define XS  (257 * 257 * 16)     // 1056784

#define PTS_PER_BLOCK 64         // 256 threads, 4 lanes per point

typedef float v4f __attribute__((ext_vector_type(4)));

__device__ __forceinline__ void acc4(v4f& a, float w, const v4f f) {
    a.x = fmaf(w, f.x, a.x);
    a.y = fmaf(w, f.y, a.y);
    a.z = fmaf(w, f.z, a.z);
    a.w = fmaf(w, f.w, a.w);
}

__global__ __launch_bounds__(256)
void FeatureVoxel_trilerp_kernel(const float* __restrict__ points,
                                 const float* __restrict__ voxel,
                                 float* __restrict__ out,
                                 int npts) {
    // Stage this block's 64 points (192 floats) into LDS with the gfx1250
    // async copy engine (tracked by ASYNCcnt), then all 8 waves read them.
    __shared__ float pts[PTS_PER_BLOCK * 3];

    const int tid = threadIdx.x;
    const int bid = blockIdx.x;

    {
        const int gidx = bid * (PTS_PER_BLOCK * 3) + tid;
        if (tid < PTS_PER_BLOCK * 3 && gidx < npts * 3) {
            // LDS flat address: low 32 bits of the generic pointer are the LDS offset.
            unsigned ldsAddr = (unsigned)(uintptr_t)(void*)pts + (unsigned)tid * 4u;
            unsigned gOff    = (unsigned)tid * 4u;
            uint64_t srcBase = (uint64_t)(uintptr_t)(points + (size_t)bid * (PTS_PER_BLOCK * 3));
            asm volatile("global_load_async_to_lds_b32 %0, %1, %2"
                         :
                         : "v"(ldsAddr), "v"(gOff), "s"(srcBase)
                         : "memory");
        }
    }
    asm volatile("s_wait_asynccnt 0" ::: "memory");
    __syncthreads();

    const int p   = tid >> 2;            // local point id, 0..63
    const int cq  = tid & 3;             // which float4 of the 16 channels
    const int pid = bid * PTS_PER_BLOCK + p;
    if (pid >= npts) return;

    const float x = pts[3 * p + 0];
    const float y = pts[3 * p + 1];
    const float z = pts[3 * p + 2];

    const int ix = (int)x, iy = (int)y, iz = (int)z;
    const float fx = x - (float)ix, fy = y - (float)iy, fz = z - (float)iz;
    const float gx = 1.0f - fx, gy = 1.0f - fy, gz = 1.0f - fz;

    // Base of corner (0,0,0), channel group cq. Each lane loads a contiguous
    // 16B slice; the 4 lanes of a point cover one full 64B corner row.
    const size_t base = (size_t)ix * XS + (size_t)iy * YS + (size_t)iz * ZS
                      + (size_t)cq * 4;
    const v4f* __restrict__ vp = (const v4f*)(voxel + base);

    // 8 corner offsets in v4f units; byte offsets max ~4.23MB < 24-bit IOFFSET,
    // so all 8 loads share one address with folded immediates. Voxel loads stay
    // RT: the 192MB L2 residual hit rate is the only reuse this kernel has.
    const v4f f000 = vp[0];
    const v4f f001 = vp[ZS / 4];
    const v4f f010 = vp[YS / 4];
    const v4f f011 = vp[(YS + ZS) / 4];
    const v4f f100 = vp[XS / 4];
    const v4f f101 = vp[(XS + ZS) / 4];
    const v4f f110 = vp[(XS + YS) / 4];
    const v4f f111 = vp[(XS + YS + ZS) / 4];

    const float gxgy = gx * gy, gxfy = gx * fy, fxgy = fx * gy, fxfy = fx * fy;

    v4f acc = (v4f)(0.0f);
    acc4(acc, gxgy * gz, f000);
    acc4(acc, gxgy * fz, f001);
    acc4(acc, gxfy * gz, f010);
    acc4(acc, gxfy * fz, f011);
    acc4(acc, fxgy * gz, f100);
    acc4(acc, fxgy * fz, f101);
    acc4(acc, fxfy * gz, f110);
    acc4(acc, fxfy * fz, f111);

    // Output (64MB) is write-once: non-temporal b128 store keeps L2 free for
    // the voxel gather stream.
    v4f* __restrict__ op = (v4f*)out;
    __builtin_nontemporal_store(acc, &op[(size_t)pid * 4 + cq]);
}

extern "C" void kernel_launch(void* const* d_in, const int* in_sizes, int n_in,
                              void* d_out, int out_size, void* d_ws, size_t ws_size,
                              hipStream_t stream) {
    const float* points = (const float*)d_in[0];   // (N, 3) f32
    const float* voxel  = (const float*)d_in[1];   // (257,257,257,16) f32
    float* out          = (float*)d_out;           // (N, 16) f32

    const int npts   = in_sizes[0] / 3;
    const int blocks = (npts + PTS_PER_BLOCK - 1) / PTS_PER_BLOCK;

    FeatureVoxel_trilerp_kernel<<<blocks, 256, 0, stream>>>(points, voxel, out, npts);
}